// SepTemporalAttention_446676599070
// MI455X (gfx1250) — compile-verified
//
#include <hip/hip_runtime.h>
#include <hip/hip_bf16.h>

// Shapes from the reference
#define B_   1024
#define T_   10
#define J_   24
#define D_   128
#define H_   8
#define DEP  16
#define R_   19        // 2*MAX_REL_POS + 1
#define XS_STRIDE 132  // 128 + 4 pad -> conflict-free row access in 64-bank LDS
#define TS   17        // 16 + 1 pad for 16x16 tiles

typedef float v2f __attribute__((ext_vector_type(2)));
typedef float v8f __attribute__((ext_vector_type(8)));

// ---------------------------------------------------------------------------
// Kernel 0: one-shot positional-encoding table (16 x 128, rows 10..15 zero).
// ---------------------------------------------------------------------------
__global__ void pe_kernel(float* __restrict__ pe)
{
    const int tid = threadIdx.x;
    for (int i = tid; i < 16 * D_; i += 256) {
        int t = i >> 7;
        int d = i & 127;
        float v = 0.0f;
        if (t < T_) {
            // angle_rate = 10000^(-2*(d//2)/128)
            float expo = -(float)(2 * (d >> 1)) * (9.210340371976184f / (float)D_);
            float rate = __expf(expo);
            float ang  = (float)t * rate;
            v = (d & 1) ? __cosf(ang) : __sinf(ang);
        }
        pe[i] = v;
    }
}

// ---------------------------------------------------------------------------
// Kernel 1: fused (x + posenc) -> q,k,v projection (WMMA f32) -> relative-
// position attention (VALU, tiny) -> concat layout write.
// One block per (b, j); wave w == head h.
// ---------------------------------------------------------------------------
__launch_bounds__(256)
__global__ void qkv_attn_kernel(const float* __restrict__ x,
                                const float* __restrict__ pe,
                                const float* __restrict__ mask,
                                const float* __restrict__ Wq, const float* __restrict__ bq,
                                const float* __restrict__ Wk, const float* __restrict__ bk,
                                const float* __restrict__ Wv, const float* __restrict__ bv,
                                const float* __restrict__ ktab, const float* __restrict__ vtab,
                                float* __restrict__ concat, float* __restrict__ attn_out)
{
    __shared__ float xs[16 * XS_STRIDE];        // x + pe, rows 10..15 zero
    __shared__ float qs[H_][16 * TS];
    __shared__ float ks[H_][16 * TS];
    __shared__ float vs[H_][16 * TS];
    __shared__ float as_[H_][16 * TS];
    __shared__ float msk[T_ * T_];
    __shared__ float kt[R_ * DEP];
    __shared__ float vt[R_ * DEP];

    const int blk  = blockIdx.x;     // b * J_ + j
    const int b    = blk / J_;
    const int j    = blk % J_;
    const int tid  = threadIdx.x;
    const int wave = tid >> 5;       // head index
    const int lane = tid & 31;
    const int l15  = lane & 15;
    const int half = lane >> 4;      // 0 or 1 (K sub-block select)

    // ---- stage x + pe into LDS with b128 loads/stores ----
    for (int i4 = tid * 4; i4 < 16 * D_; i4 += 256 * 4) {
        int t = i4 >> 7;         // row
        int d = i4 & 127;        // col (multiple of 4)
        float4 v = make_float4(0.f, 0.f, 0.f, 0.f);
        if (t < T_) {
            const float4 xv = *(const float4*)&x[(((size_t)b * T_ + t) * J_ + j) * (size_t)D_ + d];
            const float4 pv = *(const float4*)&pe[i4];
            v = make_float4(xv.x + pv.x, xv.y + pv.y, xv.z + pv.z, xv.w + pv.w);
        }
        *(float4*)&xs[t * XS_STRIDE + d] = v;   // 528B row stride: 16B aligned
    }
    for (int i = tid; i < T_ * T_; i += 256) msk[i] = mask[i];
    for (int i = tid; i < R_ * DEP; i += 256) { kt[i] = ktab[i]; vt[i] = vtab[i]; }
    __syncthreads();

    // ---- per-head GEMM: 16x16 tiles of q,k,v ; K = 128 via 32 WMMAs each ----
    const int h  = wave;
    const int nb = h * DEP;                       // output column base
    const float* Wqj = Wq + (size_t)j * D_ * D_;
    const float* Wkj = Wk + (size_t)j * D_ * D_;
    const float* Wvj = Wv + (size_t)j * D_ * D_;

    v8f qc = {}; v8f kc = {}; v8f vc = {};
    #pragma unroll
    for (int k0 = 0; k0 < D_; k0 += 4) {
        const int ka = k0 + 2 * half;
        // A (16x4): lane&15 = row, VGPR0/1 = K = ka, ka+1
        v2f a;
        a.x = xs[l15 * XS_STRIDE + ka];
        a.y = xs[l15 * XS_STRIDE + ka + 1];
        // B (4x16): lane&15 = col, VGPR0/1 = K = ka, ka+1  (coalesced b32 pairs)
        const int col = nb + l15;
        v2f bqv, bkv, bvv;
        bqv.x = Wqj[(size_t)ka * D_ + col];  bqv.y = Wqj[(size_t)(ka + 1) * D_ + col];
        bkv.x = Wkj[(size_t)ka * D_ + col];  bkv.y = Wkj[(size_t)(ka + 1) * D_ + col];
        bvv.x = Wvj[(size_t)ka * D_ + col];  bvv.y = Wvj[(size_t)(ka + 1) * D_ + col];
        qc = __builtin_amdgcn_wmma_f32_16x16x4_f32(false, a, false, bqv, (short)0, qc, false, false);
        kc = __builtin_amdgcn_wmma_f32_16x16x4_f32(false, a, false, bkv, (short)0, kc, false, false);
        vc = __builtin_amdgcn_wmma_f32_16x16x4_f32(false, a, false, bvv, (short)0, vc, false, false);
    }

    // ---- bias add + park q,k,v tiles in per-wave LDS ----
    const float biasq = bq[(size_t)j * D_ + nb + l15];
    const float biask = bk[(size_t)j * D_ + nb + l15];
    const float biasv = bv[(size_t)j * D_ + nb + l15];
    #pragma unroll
    for (int r = 0; r < 8; ++r) {
        const int row = r + 8 * half;          // C layout: VGPR r -> rows r, r+8
        qs[h][row * TS + l15] = qc[r] + biasq;
        ks[h][row * TS + l15] = kc[r] + biask;
        vs[h][row * TS + l15] = vc[r] + biasv;
    }
    __syncthreads();

    // ---- attention core (per wave; lane == query index t) ----
    if (lane < T_) {
        const int t = lane;
        float row[T_];
        float mx = -1e30f;
        #pragma unroll
        for (int s = 0; s < T_; ++s) {
            const float* krel = &kt[(s - t + 9) * DEP];
            float acc = 0.0f;
            #pragma unroll
            for (int d = 0; d < DEP; ++d)
                acc += qs[h][t * TS + d] * (ks[h][s * TS + d] + krel[d]);
            acc = acc * 0.25f + msk[t * T_ + s] * (-1e9f);   // 1/sqrt(16)
            row[s] = acc;
            mx = fmaxf(mx, acc);
        }
        float sum = 0.0f;
        #pragma unroll
        for (int s = 0; s < T_; ++s) { row[s] = __expf(row[s] - mx); sum += row[s]; }
        const float inv = 1.0f / sum;
        #pragma unroll
        for (int s = 0; s < T_; ++s) as_[h][t * TS + s] = row[s] * inv;
    }
    __syncthreads();

    // last-query attention weights -> second output
    if (lane < T_) {
        attn_out[(((size_t)b * J_ + j) * H_ + h) * T_ + lane] = as_[h][9 * TS + lane];
    }

    // out = attn @ (v + rel_val), written directly in concat (B,T,J,D) layout
    for (int i = lane; i < T_ * DEP; i += 32) {
        const int t = i >> 4;
        const int d = i & 15;
        float acc = 0.0f;
        #pragma unroll
        for (int s = 0; s < T_; ++s)
            acc += as_[h][t * TS + s] * (vs[h][s * TS + d] + vt[(s - t + 9) * DEP + d]);
        concat[(((size_t)b * T_ + t) * J_ + j) * (size_t)D_ + nb + d] = acc;
    }
}

// ---------------------------------------------------------------------------
// Kernel 2: output projection  out = concat @ Wo + bo
// One block per 16-row A tile (M = B*T*J = 245760); 8 waves cover N = 128.
// ---------------------------------------------------------------------------
__launch_bounds__(256)
__global__ void proj_kernel(const float* __restrict__ concat,
                            const float* __restrict__ Wo, const float* __restrict__ bo,
                            float* __restrict__ out)
{
    __shared__ float asx[16 * XS_STRIDE];

    const int mb   = blockIdx.x * 16;
    const int tid  = threadIdx.x;
    const int wave = tid >> 5;
    const int lane = tid & 31;
    const int l15  = lane & 15;
    const int half = lane >> 4;

    // stage A tile (16 x 128) with b128 loads -- fully coalesced
    for (int i4 = tid * 4; i4 < 16 * D_; i4 += 256 * 4) {
        int r = i4 >> 7;
        int d = i4 & 127;
        *(float4*)&asx[r * XS_STRIDE + d] =
            *(const float4*)&concat[(size_t)(mb + r) * D_ + d];
    }
    __syncthreads();

    const int nbb = wave * 16;
    v8f c = {};
    #pragma unroll
    for (int k0 = 0; k0 < D_; k0 += 4) {
        const int ka = k0 + 2 * half;
        v2f a;
        a.x = asx[l15 * XS_STRIDE + ka];
        a.y = asx[l15 * XS_STRIDE + ka + 1];
        v2f bb;
        bb.x = Wo[(size_t)ka * D_ + nbb + l15];
        bb.y = Wo[(size_t)(ka + 1) * D_ + nbb + l15];
        c = __builtin_amdgcn_wmma_f32_16x16x4_f32(false, a, false, bb, (short)0, c, false, false);
    }

    const float bias = bo[nbb + l15];
    #pragma unroll
    for (int r = 0; r < 8; ++r) {
        const int row = r + 8 * half;
        out[(size_t)(mb + row) * D_ + nbb + l15] = c[r] + bias;
    }
}

// ---------------------------------------------------------------------------
extern "C" void kernel_launch(void* const* d_in, const int* in_sizes, int n_in,
                              void* d_out, int out_size, void* d_ws, size_t ws_size,
                              hipStream_t stream)
{
    const float* x    = (const float*)d_in[0];
    const float* mask = (const float*)d_in[1];
    const float* Wq   = (const float*)d_in[2];
    const float* bq   = (const float*)d_in[3];
    const float* Wk   = (const float*)d_in[4];
    const float* bk   = (const float*)d_in[5];
    const float* Wv   = (const float*)d_in[6];
    const float* bv   = (const float*)d_in[7];
    const float* Wo   = (const float*)d_in[8];
    const float* bo   = (const float*)d_in[9];
    const float* ktab = (const float*)d_in[10];
    const float* vtab = (const float*)d_in[11];

    float* outputs = (float*)d_out;                               // (B,T,J,D)
    float* attn_w  = outputs + (size_t)B_ * T_ * J_ * D_;         // (B,J,H,T)

    float* pe      = (float*)d_ws;                                // 16*128 floats
    float* concat  = pe + 16 * D_;                                // (B,T,J,D) scratch

    pe_kernel<<<1, 256, 0, stream>>>(pe);
    qkv_attn_kernel<<<B_ * J_, 256, 0, stream>>>(x, pe, mask, Wq, bq, Wk, bk, Wv, bv,
                                                 ktab, vtab, concat, attn_w);
    proj_kernel<<<(B_ * T_ * J_) / 16, 256, 0, stream>>>(concat, Wo, bo, outputs);
}